// LinearTPReadOut_3839700763111
// MI455X (gfx1250) — compile-verified
//
#include <hip/hip_runtime.h>

typedef float v2f __attribute__((ext_vector_type(2)));
typedef float v8f __attribute__((ext_vector_type(8)));

#define ROW_STRIDE 1152   // x columns: 128 + 384 + 640
#define SLICE_OFF  128
#define SLICE_DIM  384    // 128 u * 3 i, contiguous in memory
#define NCHUNK     96     // 384 / 4 K-chunks
#define WAVES_PER_BLOCK 4

// One wave32 computes out[n0 .. n0+15] via D(16x16) += A(16x4)*B(4x16).
// A = x-slice rows (one b64 load/lane/chunk), B = structured weights,
// pre-baked per-lane into LDS in the exact WMMA operand layout.
__global__ void lin_tp_readout_wmma(const float* __restrict__ x,
                                    const float* __restrict__ w_lin,
                                    const float* __restrict__ w_tp,
                                    float* __restrict__ out, int n) {
  // B~[k, c] = (k%3 == c%3 && c < 6) ? w_lin[k/3, c/3] : 0, c = 3w + i.
  // Stored as the per-lane v2f each lane feeds to chunk j:
  //   half = lane>>4, lp = lane&15, kA = 4j + 2*half  ->  {B~[kA,lp], B~[kA+1,lp]}
  __shared__ v2f sB[NCHUNK * 32];                       // 24 KB, block-shared

  for (int idx = threadIdx.x; idx < NCHUNK * 32; idx += blockDim.x) {
    const int j    = idx >> 5;
    const int ln   = idx & 31;
    const int lp   = ln & 15;
    const int kA   = 4 * j + 2 * (ln >> 4);
    const int i_c  = (lp < 6) ? (lp % 3) : 3;           // 3 never matches k%3
    const int w_c  = (lp < 6) ? (lp / 3) : 0;
    v2f b;
    b.x = ((kA)     % 3 == i_c) ? w_lin[((kA)     / 3) * 2 + w_c] : 0.0f;
    b.y = ((kA + 1) % 3 == i_c) ? w_lin[((kA + 1) / 3) * 2 + w_c] : 0.0f;
    sB[idx] = b;
  }
  __syncthreads();

  const int lane = threadIdx.x & 31;
  const int wave = threadIdx.x >> 5;
  const int tile = blockIdx.x * WAVES_PER_BLOCK + wave;
  const int n0   = tile * 16;
  if (n0 >= n) return;                                  // wave-uniform exit

  const int lp   = lane & 15;                           // A-row in tile
  const int half = lane >> 4;                           // K pair 0,1 vs 2,3

  int row = n0 + lp; if (row >= n) row = n - 1;         // clamp tail loads
  const float* xrow = x + (size_t)row * ROW_STRIDE + SLICE_OFF + 2 * half;
  const v2f* bcol = sB + lane;                          // + j*32, imm offsets

  v8f acc = {0.f, 0.f, 0.f, 0.f, 0.f, 0.f, 0.f, 0.f};

  #pragma unroll 8
  for (int j = 0; j < NCHUNK; ++j) {
    if ((j & 7) == 0)                                   // 128B-line lookahead
      __builtin_prefetch(xrow + 4 * j + 256, 0, 1);
    v2f a = *(const v2f*)(xrow + 4 * j);                // A: K(2h), K(2h+1)
    v2f b = bcol[j * 32];                               // B: same K split
    acc = __builtin_amdgcn_wmma_f32_16x16x4_f32(
        /*neg_a=*/false, a, /*neg_b=*/false, b,
        /*c_mod=*/(short)0, acc, /*reuse_a=*/false, /*reuse_b=*/false);
  }

  // out[n] = w_tp/(128*sqrt(3)) * sum_i D[r, i]*D[r, 3+i]
  const float scale = w_tp[0] * (1.0f / (128.0f * 1.7320508075688772f));
  const int   base  = lane & 16;                        // half owns rows r / r+8

  #pragma unroll
  for (int r = 0; r < 8; ++r) {
    float v  = acc[r];                                  // D[row, lp]
    float u  = __shfl(v, base + ((lp + 3) & 15), 32);   // D[row, lp+3]
    float p  = v * u;                                   // a_i*b_i at lanes 0..2
    float p1 = __shfl(p, base + ((lp + 1) & 15), 32);
    float p2 = __shfl(p, base + ((lp + 2) & 15), 32);
    if (lp == 0) {
      int nn = n0 + r + (half ? 8 : 0);
      if (nn < n) out[nn] = scale * (p + p1 + p2);
    }
  }
}

extern "C" void kernel_launch(void* const* d_in, const int* in_sizes, int n_in,
                              void* d_out, int out_size, void* d_ws, size_t ws_size,
                              hipStream_t stream) {
  const float* x     = (const float*)d_in[0];
  const float* w_lin = (const float*)d_in[1];
  const float* w_tp  = (const float*)d_in[2];
  float* out = (float*)d_out;

  const int n      = in_sizes[0] / ROW_STRIDE;          // 200000
  const int tiles  = (n + 15) / 16;                     // 12500 wave-tiles
  const int blocks = (tiles + WAVES_PER_BLOCK - 1) / WAVES_PER_BLOCK;

  lin_tp_readout_wmma<<<blocks, 32 * WAVES_PER_BLOCK, 0, stream>>>(
      x, w_lin, w_tp, out, n);
}